// GraphSAGEModel_66992899883186
// MI455X (gfx1250) — compile-verified
//
#include <hip/hip_runtime.h>

// GraphSAGE link-prediction pipeline for gfx1250 (MI455X), fp32 throughout.
// Matrix work uses V_WMMA_F32_16X16X4_F32 (exact fp32 on the matrix pipe).

#define NNODES 50000
#define HID    128
#define NEDGES 800000
#define NQUERY 200000

// 128 + 4 pad: A-fragment LDS reads (16 lanes, row stride 132 words -> banks
// m*4 mod 64, upper half-wave +2) are conflict-free.
#define LDSTRIDE 132

typedef __attribute__((ext_vector_type(2))) float v2f;
typedef __attribute__((ext_vector_type(8))) float v8f;

// ---------------------------------------------------------------- utilities
__global__ __launch_bounds__(256) void zero_f4(float* __restrict__ p, int n4) {
  int i = blockIdx.x * 256 + threadIdx.x;
  if (i < n4) ((float4*)p)[i] = make_float4(0.f, 0.f, 0.f, 0.f);
}

__global__ __launch_bounds__(256) void deg_kernel(const int* __restrict__ dst,
                                                  float* __restrict__ deg, int e) {
  int i = blockIdx.x * 256 + threadIdx.x;
  if (i < e) atomicAdd(&deg[dst[i]], 1.0f);
}

__global__ __launch_bounds__(256) void rdeg_kernel(const float* __restrict__ deg,
                                                   float* __restrict__ rdeg, int n) {
  int i = blockIdx.x * 256 + threadIdx.x;
  if (i < n) rdeg[i] = 1.0f / fmaxf(deg[i], 1.0f);
}

// scatter-add of neighbor features: agg[dst] += zin[src].
// one thread per (edge, 4 features); L2-resident destination, f32 atomics.
__global__ __launch_bounds__(256) void scatter_kernel(const float* __restrict__ zin,
                                                      const int* __restrict__ src,
                                                      const int* __restrict__ dst,
                                                      float* __restrict__ agg, int e) {
  int i = blockIdx.x * 256 + threadIdx.x;
  int eidx = i >> 5;
  int c4   = i & 31;
  if (eidx < e) {
    int s = src[eidx];
    int d = dst[eidx];
    float4 v = ((const float4*)(zin + (size_t)s * HID))[c4];
    float* a = agg + (size_t)d * HID + (c4 << 2);
    atomicAdd(a + 0, v.x);
    atomicAdd(a + 1, v.y);
    atomicAdd(a + 2, v.z);
    atomicAdd(a + 3, v.w);
  }
}

// ------------------------------------------------------------ WMMA K=128 pass
// acc[t] += A_tile(t) x W^T  for 4 row tiles, one 16-col tile per wave.
// A in LDS (padded), W rows from global (L2-resident, 64 KB).
__device__ __forceinline__ void wmma_pass(const float (*A)[LDSTRIDE],
                                          const float* __restrict__ Wrow,
                                          int nl, int ksel, v8f acc[4]) {
  __builtin_prefetch(Wrow, 0, 1);  // global_prefetch_b8
#pragma unroll 4
  for (int k = 0; k < HID; k += 4) {
    v2f b;
    b.x = Wrow[k + ksel];
    b.y = Wrow[k + ksel + 1];
#pragma unroll
    for (int t = 0; t < 4; ++t) {
      v2f a;
      a.x = A[t * 16 + nl][k + ksel];
      a.y = A[t * 16 + nl][k + ksel + 1];
      acc[t] = __builtin_amdgcn_wmma_f32_16x16x4_f32(
          /*neg_a=*/false, a, /*neg_b=*/false, b,
          /*c_mod=*/(short)0, acc[t], /*reuse_a=*/false, /*reuse_b=*/false);
    }
  }
}

// ---------------------------------------------------- SAGE linear (fused)
// out[n] = maybe_relu( (agg[n]*rdeg[n]) @ Wl^T + bl + zin[n] @ Wr^T )
__global__ __launch_bounds__(256) void sage_linear_kernel(
    const float* __restrict__ agg, const float* __restrict__ rdeg,
    const float* __restrict__ zin, const float* __restrict__ Wl,
    const float* __restrict__ bl, const float* __restrict__ Wr,
    float* __restrict__ out, int do_relu) {
  __shared__ float sA[64][LDSTRIDE];
  const int tid  = threadIdx.x;
  const int lane = tid & 31;
  const int wave = tid >> 5;
  const int row0 = blockIdx.x * 64;
  const int nl   = lane & 15;
  const int ksel = (lane >> 4) << 1;  // 0 or 2
  const int n0   = wave * 16;

  // stage mean = agg * rdeg (coalesced float4)
  for (int i = tid; i < 64 * 32; i += 256) {
    int r = i >> 5, c4 = i & 31;
    int node = row0 + r;
    float4 m = make_float4(0.f, 0.f, 0.f, 0.f);
    if (node < NNODES) {
      float s  = rdeg[node];
      float4 a = ((const float4*)(agg + (size_t)node * HID))[c4];
      m = make_float4(a.x * s, a.y * s, a.z * s, a.w * s);
    }
    *(float4*)&sA[r][c4 << 2] = m;
  }
  __syncthreads();

  v8f acc[4];
#pragma unroll
  for (int t = 0; t < 4; ++t)
#pragma unroll
    for (int j = 0; j < 8; ++j) acc[t][j] = 0.f;

  wmma_pass(sA, Wl + (size_t)(n0 + nl) * HID, nl, ksel, acc);
  __syncthreads();

  // restage root features into the same buffer, continue accumulation
  for (int i = tid; i < 64 * 32; i += 256) {
    int r = i >> 5, c4 = i & 31;
    int node = row0 + r;
    float4 zv = make_float4(0.f, 0.f, 0.f, 0.f);
    if (node < NNODES) zv = ((const float4*)(zin + (size_t)node * HID))[c4];
    *(float4*)&sA[r][c4 << 2] = zv;
  }
  __syncthreads();

  wmma_pass(sA, Wr + (size_t)(n0 + nl) * HID, nl, ksel, acc);

  const float bias = bl[n0 + nl];
#pragma unroll
  for (int t = 0; t < 4; ++t) {
    int mbase = t * 16 + ((lane >> 4) << 3);
#pragma unroll
    for (int r = 0; r < 8; ++r) {
      int node = row0 + mbase + r;
      if (node < NNODES) {
        float v = acc[t][r] + bias;
        if (do_relu) v = fmaxf(v, 0.f);
        out[(size_t)node * HID + n0 + nl] = v;
      }
    }
  }
}

// ------------------------------------------------------- fused decoder MLP
// h = z[qs]*z[qd]; h = relu(h@Wd1^T+bd1); h = relu(h@Wd2^T+bd2);
// logits = h@Wd3^T + bd3   (all in one block per 64 queries, LDS-resident)
__global__ __launch_bounds__(256) void decoder_kernel(
    const float* __restrict__ z, const int* __restrict__ qs,
    const int* __restrict__ qd, const float* __restrict__ Wd1,
    const float* __restrict__ bd1, const float* __restrict__ Wd2,
    const float* __restrict__ bd2, const float* __restrict__ Wd3,
    const float* __restrict__ bd3, float* __restrict__ out) {
  __shared__ float sH[64][LDSTRIDE];
  const int tid  = threadIdx.x;
  const int lane = tid & 31;
  const int wave = tid >> 5;
  const int q0   = blockIdx.x * 64;
  const int nl   = lane & 15;
  const int ksel = (lane >> 4) << 1;
  const int n0   = wave * 16;
  const int mofs = (lane >> 4) << 3;

  // gather h0 = z[qs] * z[qd]
  for (int i = tid; i < 64 * 32; i += 256) {
    int r = i >> 5, c4 = i & 31;
    int q = q0 + r;
    float4 h = make_float4(0.f, 0.f, 0.f, 0.f);
    if (q < NQUERY) {
      float4 a = ((const float4*)(z + (size_t)qs[q] * HID))[c4];
      float4 b = ((const float4*)(z + (size_t)qd[q] * HID))[c4];
      h = make_float4(a.x * b.x, a.y * b.y, a.z * b.z, a.w * b.w);
    }
    *(float4*)&sH[r][c4 << 2] = h;
  }
  __syncthreads();

  v8f acc[4];

  // ---- layer 1
#pragma unroll
  for (int t = 0; t < 4; ++t)
#pragma unroll
    for (int j = 0; j < 8; ++j) acc[t][j] = 0.f;
  wmma_pass(sH, Wd1 + (size_t)(n0 + nl) * HID, nl, ksel, acc);
  __syncthreads();
  {
    float bias = bd1[n0 + nl];
#pragma unroll
    for (int t = 0; t < 4; ++t)
#pragma unroll
      for (int r = 0; r < 8; ++r)
        sH[t * 16 + mofs + r][n0 + nl] = fmaxf(acc[t][r] + bias, 0.f);
  }
  __syncthreads();

  // ---- layer 2
#pragma unroll
  for (int t = 0; t < 4; ++t)
#pragma unroll
    for (int j = 0; j < 8; ++j) acc[t][j] = 0.f;
  wmma_pass(sH, Wd2 + (size_t)(n0 + nl) * HID, nl, ksel, acc);
  __syncthreads();
  {
    float bias = bd2[n0 + nl];
#pragma unroll
    for (int t = 0; t < 4; ++t)
#pragma unroll
      for (int r = 0; r < 8; ++r)
        sH[t * 16 + mofs + r][n0 + nl] = fmaxf(acc[t][r] + bias, 0.f);
  }
  __syncthreads();

  // ---- head: 128 -> 2 (64 rows x 2 cols = 128 dot products)
  if (tid < 128) {
    int r = tid >> 1, c = tid & 1;
    int q = q0 + r;
    if (q < NQUERY) {
      const float* w = Wd3 + c * HID;
      float s = bd3[c];
#pragma unroll 8
      for (int k = 0; k < HID; ++k) s += sH[r][k] * w[k];
      out[(size_t)q * 2 + c] = s;
    }
  }
}

// ------------------------------------------------------------------- launch
extern "C" void kernel_launch(void* const* d_in, const int* in_sizes, int n_in,
                              void* d_out, int out_size, void* d_ws,
                              size_t ws_size, hipStream_t stream) {
  (void)in_sizes; (void)n_in; (void)out_size; (void)ws_size;

  const float* x   = (const float*)d_in[0];
  const int*   ei  = (const int*)d_in[1];
  const int*   eli = (const int*)d_in[2];
  const float* Wl1 = (const float*)d_in[3];
  const float* bl1 = (const float*)d_in[4];
  const float* Wr1 = (const float*)d_in[5];
  const float* Wl2 = (const float*)d_in[6];
  const float* bl2 = (const float*)d_in[7];
  const float* Wr2 = (const float*)d_in[8];
  const float* Wl3 = (const float*)d_in[9];
  const float* bl3 = (const float*)d_in[10];
  const float* Wr3 = (const float*)d_in[11];
  const float* Wd1 = (const float*)d_in[12];
  const float* bd1 = (const float*)d_in[13];
  const float* Wd2 = (const float*)d_in[14];
  const float* bd2 = (const float*)d_in[15];
  const float* Wd3 = (const float*)d_in[16];
  const float* bd3 = (const float*)d_in[17];
  float* out = (float*)d_out;

  // workspace layout (floats): deg[N], rdeg[N], agg[N*128], zA[N*128], zB[N*128]
  float* ws   = (float*)d_ws;
  float* deg  = ws;
  float* rdeg = ws + NNODES;
  float* agg  = ws + 2 * NNODES;
  float* zA   = agg + (size_t)NNODES * HID;
  float* zB   = zA + (size_t)NNODES * HID;

  const int* src = ei;
  const int* dst = ei + NEDGES;
  const int* qsp = eli;
  const int* qdp = eli + NQUERY;

  const int featN4   = NNODES * HID / 4;                 // 1.6M float4
  const int zeroBlkF = (featN4 + 255) / 256;
  const int degN4    = NNODES / 4;
  const int scatBlk  = (NEDGES * 32 + 255) / 256;
  const int sageBlk  = (NNODES + 63) / 64;               // 782
  const int decBlk   = (NQUERY + 63) / 64;               // 3125

  // degrees (graph-constant across layers)
  zero_f4<<<(degN4 + 255) / 256, 256, 0, stream>>>(deg, degN4);
  deg_kernel<<<(NEDGES + 255) / 256, 256, 0, stream>>>(dst, deg, NEDGES);
  rdeg_kernel<<<(NNODES + 255) / 256, 256, 0, stream>>>(deg, rdeg, NNODES);

  // layer 1: x -> zA (relu)
  zero_f4<<<zeroBlkF, 256, 0, stream>>>(agg, featN4);
  scatter_kernel<<<scatBlk, 256, 0, stream>>>(x, src, dst, agg, NEDGES);
  sage_linear_kernel<<<sageBlk, 256, 0, stream>>>(agg, rdeg, x, Wl1, bl1, Wr1,
                                                  zA, 1);
  // layer 2: zA -> zB (relu)
  zero_f4<<<zeroBlkF, 256, 0, stream>>>(agg, featN4);
  scatter_kernel<<<scatBlk, 256, 0, stream>>>(zA, src, dst, agg, NEDGES);
  sage_linear_kernel<<<sageBlk, 256, 0, stream>>>(agg, rdeg, zA, Wl2, bl2, Wr2,
                                                  zB, 1);
  // layer 3: zB -> zA (no relu)
  zero_f4<<<zeroBlkF, 256, 0, stream>>>(agg, featN4);
  scatter_kernel<<<scatBlk, 256, 0, stream>>>(zB, src, dst, agg, NEDGES);
  sage_linear_kernel<<<sageBlk, 256, 0, stream>>>(agg, rdeg, zB, Wl3, bl3, Wr3,
                                                  zA, 0);

  // fused decoder -> logits
  decoder_kernel<<<decBlk, 256, 0, stream>>>(zA, qsp, qdp, Wd1, bd1, Wd2, bd2,
                                             Wd3, bd3, out);
}